// PointNGCF_70944269795902
// MI455X (gfx1250) — compile-verified
//
#include <hip/hip_runtime.h>

#define N_USER_  50000
#define N_ITEM_  50000
#define NNODES   (N_USER_ + N_ITEM_)
#define DIM      64
#define NLAYERS  3
#define BATCH_   4096

typedef __attribute__((ext_vector_type(2))) float v2f;
typedef __attribute__((ext_vector_type(8))) float v8f;

// ---------------------------------------------------------------------------
// ego = concat(user_emb, item_emb); also layer-0 slab of all_e [N, 4*D]
// ---------------------------------------------------------------------------
__global__ void init_ego_kernel(const float* __restrict__ ue,
                                const float* __restrict__ ie,
                                float* __restrict__ ego,
                                float* __restrict__ all_e) {
  int idx = blockIdx.x * blockDim.x + threadIdx.x;
  if (idx >= NNODES * DIM) return;
  int node = idx >> 6;
  int d = idx & 63;
  float v = (node < N_USER_) ? ue[idx] : ie[(size_t)(node - N_USER_) * DIM + d];
  ego[idx] = v;
  all_e[(size_t)node * (4 * DIM) + d] = v;
}

// ---------------------------------------------------------------------------
// COO SpMM scatter: side[row] += val * ego[col]
// one thread per (edge, 4-dim chunk): float4 gather + 4 f32 atomic scatters
// ---------------------------------------------------------------------------
__global__ void spmm_kernel(const int* __restrict__ row,
                            const int* __restrict__ col,
                            const float* __restrict__ val,
                            const float* __restrict__ ego,
                            float* __restrict__ side, int nnz) {
  unsigned gid = blockIdx.x * blockDim.x + threadIdx.x;
  unsigned e = gid >> 4;
  unsigned q = gid & 15u;
  if (e >= (unsigned)nnz) return;
  int r = row[e], c = col[e];
  float v = val[e];
  const float4 x = ((const float4*)(ego + (size_t)c * DIM))[q];
  float* dst = side + (size_t)r * DIM + q * 4;
  atomicAdd(dst + 0, v * x.x);
  atomicAdd(dst + 1, v * x.y);
  atomicAdd(dst + 2, v * x.z);
  atomicAdd(dst + 3, v * x.w);
}

// ---------------------------------------------------------------------------
// Fused layer: ego_new = leaky_relu(side@Wgc + bgc + (ego*side)@Wbi + bbi)
//              all_e[:, (k+1)*64:(k+2)*64] = l2norm(ego_new)
// One wave32 per 16-row x 64-col tile; both GEMMs accumulate into the same
// v_wmma_f32_16x16x4_f32 accumulators. W staged in LDS, K-pair interleaved
// so every B fragment is a single aligned ds_load_b64.
// ---------------------------------------------------------------------------
__global__ __launch_bounds__(256) void layer_kernel(
    const float* __restrict__ side, float* __restrict__ ego,
    const float* __restrict__ Wgc, const float* __restrict__ bgc,
    const float* __restrict__ Wbi, const float* __restrict__ bbi,
    float* __restrict__ all_e, int layer) {
  // sW[(k/2)*128 + 2n + (k&1)] = W[k*64 + n]  -> frag(kpair,n) is contiguous
  __shared__ __align__(16) float sWgc[DIM * DIM];
  __shared__ __align__(16) float sWbi[DIM * DIM];
  for (int i = threadIdx.x; i < DIM * DIM; i += 256) {
    const int k = i >> 6, n = i & 63;
    const int s = (k >> 1) * 128 + 2 * n + (k & 1);
    sWgc[s] = Wgc[i];
    sWbi[s] = Wbi[i];
  }
  __syncthreads();

  const int tid = threadIdx.x;
  const int wid = tid >> 5;       // 8 waves per block
  const int lane = tid & 31;
  const int hi = lane >> 4;       // which 16-lane half
  const int r = lane & 15;
  const int rowBase = blockIdx.x * 128 + wid * 16;
  const int arow = rowBase + r;
  // branch-free bounds handling: clamp address, zero contribution via weight
  const float w = (arow < NNODES) ? 1.f : 0.f;
  const int lrow = (arow < NNODES) ? arow : (NNODES - 1);
  const float* srow = side + (size_t)lrow * DIM;
  const float* erow = ego + (size_t)lrow * DIM;

  v8f acc[4] = {};                // 4 column tiles of 16, f32 C/D

  for (int k0 = 0; k0 < DIM; k0 += 4) {
    const int kk = k0 + 2 * hi;   // A frag: lanes 0-15 hold K=k0,k0+1; 16-31 hold k0+2,k0+3
    const float s0 = srow[kk] * w;
    const float s1 = srow[kk + 1] * w;
    const float e0 = erow[kk];
    const float e1 = erow[kk + 1];
    v2f aS; aS.x = s0;      aS.y = s1;        // A for side@Wgc
    v2f aE; aE.x = s0 * e0; aE.y = s1 * e1;   // A for (ego*side)@Wbi  (s==0 if OOB)
    const int pb = ((k0 >> 1) + hi) * 128 + 2 * r;  // pair-row base in LDS
#pragma unroll
    for (int ct = 0; ct < 4; ++ct) {
      const v2f bG = *(const v2f*)&sWgc[pb + ct * 32];
      const v2f bB = *(const v2f*)&sWbi[pb + ct * 32];
      acc[ct] = __builtin_amdgcn_wmma_f32_16x16x4_f32(
          false, aS, false, bG, (short)0, acc[ct], false, false);
      acc[ct] = __builtin_amdgcn_wmma_f32_16x16x4_f32(
          false, aE, false, bB, (short)0, acc[ct], false, false);
    }
  }

  // bias + leaky_relu(0.2)
  float x[4][8];
#pragma unroll
  for (int ct = 0; ct < 4; ++ct) {
    const int n = ct * 16 + r;
    const float bias = bgc[n] + bbi[n];
#pragma unroll
    for (int i = 0; i < 8; ++i) {
      float t = acc[ct][i] + bias;
      x[ct][i] = (t > 0.f) ? t : 0.2f * t;
    }
  }

  // per-row sum of squares: reduce across the 16 lanes of each half-wave
  float p[8];
#pragma unroll
  for (int i = 0; i < 8; ++i) {
    float s = 0.f;
#pragma unroll
    for (int ct = 0; ct < 4; ++ct) s += x[ct][i] * x[ct][i];
#pragma unroll
    for (int m = 1; m < 16; m <<= 1) s += __shfl_xor(s, m, 16);
    p[i] = s;
  }

#pragma unroll
  for (int i = 0; i < 8; ++i) {
    const int row = rowBase + i + 8 * hi;     // C/D layout: M = i + 8*hi
    if (row < NNODES) {
      const float nrm = fmaxf(sqrtf(p[i]), 1e-12f);
#pragma unroll
      for (int ct = 0; ct < 4; ++ct) {
        const int cidx = ct * 16 + r;
        ego[(size_t)row * DIM + cidx] = x[ct][i];
        all_e[(size_t)row * (4 * DIM) + (size_t)(layer + 1) * DIM + cidx] =
            x[ct][i] / nrm;
      }
    }
  }
}

// ---------------------------------------------------------------------------
// Final: u_g = all_e[user], i_g = all_e[n_user + item], pos = dot(u_g, i_g)
// out = [u_g (B x 256) | i_g (B x 256) | pos (B)]
// ---------------------------------------------------------------------------
__global__ __launch_bounds__(256) void gather_kernel(
    const int* __restrict__ user, const int* __restrict__ item,
    const float* __restrict__ all_e, float* __restrict__ out) {
  const int b = blockIdx.x;
  const int t = threadIdx.x;    // 256 threads = 256 cols
  const int u = user[b];
  const int it = item[b];
  const float uv = all_e[(size_t)u * 256 + t];
  const float iv = all_e[(size_t)(N_USER_ + it) * 256 + t];
  out[(size_t)b * 256 + t] = uv;
  out[(size_t)BATCH_ * 256 + (size_t)b * 256 + t] = iv;

  // wave32 reduction, then combine 8 wave partials through LDS
  float s = uv * iv;
#pragma unroll
  for (int m = 1; m < 32; m <<= 1) s += __shfl_xor(s, m, 32);
  __shared__ float red[8];
  if ((t & 31) == 0) red[t >> 5] = s;
  __syncthreads();
  if (t == 0) {
    float tot = 0.f;
#pragma unroll
    for (int wv = 0; wv < 8; ++wv) tot += red[wv];
    out[(size_t)2 * BATCH_ * 256 + b] = tot;
  }
}

// ---------------------------------------------------------------------------
extern "C" void kernel_launch(void* const* d_in, const int* in_sizes, int n_in,
                              void* d_out, int out_size, void* d_ws, size_t ws_size,
                              hipStream_t stream) {
  const int*   user    = (const int*)d_in[0];
  const int*   item_i  = (const int*)d_in[1];
  const int*   adj_row = (const int*)d_in[2];
  const int*   adj_col = (const int*)d_in[3];
  const float* adj_val = (const float*)d_in[4];
  const float* ue      = (const float*)d_in[5];
  const float* ie      = (const float*)d_in[6];
  const float* Wgc     = (const float*)d_in[7];
  const float* bgc     = (const float*)d_in[8];
  const float* Wbi     = (const float*)d_in[9];
  const float* bbi     = (const float*)d_in[10];
  float* out = (float*)d_out;
  const int nnz = in_sizes[2];

  float* ws    = (float*)d_ws;
  float* all_e = ws;                              // NNODES * 256
  float* ego   = all_e + (size_t)NNODES * 256;    // NNODES * 64
  float* side  = ego + (size_t)NNODES * DIM;      // NNODES * 64

  init_ego_kernel<<<(NNODES * DIM + 255) / 256, 256, 0, stream>>>(ue, ie, ego, all_e);

  for (int k = 0; k < NLAYERS; ++k) {
    hipMemsetAsync(side, 0, (size_t)NNODES * DIM * sizeof(float), stream);
    unsigned total = (unsigned)nnz * 16u;
    spmm_kernel<<<(total + 255) / 256, 256, 0, stream>>>(adj_row, adj_col, adj_val,
                                                         ego, side, nnz);
    layer_kernel<<<(NNODES + 127) / 128, 256, 0, stream>>>(
        side, ego,
        Wgc + (size_t)k * DIM * DIM, bgc + (size_t)k * DIM,
        Wbi + (size_t)k * DIM * DIM, bbi + (size_t)k * DIM,
        all_e, k);
  }

  gather_kernel<<<BATCH_, 256, 0, stream>>>(user, item_i, all_e, out);
}